// YOLOLayer_11012296147525
// MI455X (gfx1250) — compile-verified
//
#include <hip/hip_runtime.h>
#include <math.h>

// ---------------------------------------------------------------------------
// YOLO layer loss + activation output for MI455X (gfx1250, wave32).
// Memory-bound: ~83MB read + ~83MB write => ~7.1us roofline @ 23.3 TB/s.
// Structure: one fused streaming kernel (activations + default noobj BCE
// partial sums), a 128-thread sparse-correction kernel (which also zeroes the
// reduction padding), and a WMMA-based deterministic final reduction
// (V_WMMA_F32_16X16X4_F32 with a ones B-matrix sums 64 partials per issue;
// the total is invariant to the A-operand lane layout, so we can use a
// contiguous float2-per-lane mapping => one global_load_b64 per lane per
// iteration, no bounds guards, no EXEC toggling in the hot loop).
// ---------------------------------------------------------------------------

#define NB 128
#define NG 52
#define NA 3
#define NC 15
#define PIX (NG * NG)                  // 2704
#define CELLS_PER_B (NA * PIX)         // 8112
#define TOTAL_CELLS (NB * CELLS_PER_B) // 1038336
#define BLK 256
#define NBLK (TOTAL_CELLS / BLK)       // 4056 (exact)
#define NPART (NBLK + 1)               // 4057 live partials
#define NPAD 4096                      // padded to a multiple of 64

__device__ __forceinline__ float sigmoidf_(float v) {
    return 1.0f / (1.0f + expf(-v));
}

// ---------------------------------------------------------------------------
// Kernel 1: fused streaming pass. One thread per (b,a,j,i) cell.
//  - 20 channel loads, stride PIX floats (each coalesced across the wave)
//  - sigmoid x/y/conf, softmax over 15 classes
//  - write 20 contiguous floats of `out`
//  - accumulate default noobj conf-BCE term; block-tree-reduce -> partial[blk]
// ---------------------------------------------------------------------------
__global__ void yolo_main(const float* __restrict__ x,
                          float* __restrict__ out,
                          float* __restrict__ partial) {
    __shared__ float red[BLK];
    const int t = blockIdx.x * BLK + threadIdx.x;   // global cell id
    const int b = t / CELLS_PER_B;
    const int r = t - b * CELLS_PER_B;
    const int a = r / PIX;
    const int p = r - a * PIX;                      // j*NG + i

    const float* base = x + (size_t)(b * 60 + a * 20) * PIX + p;

    // CDNA5 VMEM prefetch of the next block's line for this channel stream
    __builtin_prefetch(base + BLK, 0, 1);           // -> global_prefetch_b8

    float v[20];
#pragma unroll
    for (int c = 0; c < 20; ++c) v[c] = base[(size_t)c * PIX];

    const float px   = sigmoidf_(v[0]);
    const float py   = sigmoidf_(v[1]);
    const float pw   = v[2];
    const float ph   = v[3];
    const float conf = sigmoidf_(v[4]);

    // softmax over classes (max-subtracted, matches jax.nn.softmax)
    float m = v[5];
#pragma unroll
    for (int c = 6; c < 20; ++c) m = fmaxf(m, v[c]);
    float e[NC];
    float s = 0.0f;
#pragma unroll
    for (int c = 0; c < NC; ++c) { e[c] = expf(v[5 + c] - m); s += e[c]; }
    const float inv = 1.0f / s;

    float* o = out + 1 + (size_t)t * 20;            // out starts after loss scalar
    o[0] = px * 8.0f;
    o[1] = py * 8.0f;
    o[2] = pw * 8.0f;
    o[3] = ph * 8.0f;
    o[4] = conf;
#pragma unroll
    for (int c = 0; c < NC; ++c) o[5 + c] = e[c] * inv;

    // default noobj contribution: 100 * -(max(log(1-conf), -100))
    float term = -100.0f * fmaxf(logf(1.0f - conf), -100.0f);

    red[threadIdx.x] = term;
    __syncthreads();
#pragma unroll
    for (int sft = BLK / 2; sft > 0; sft >>= 1) {
        if (threadIdx.x < sft) red[threadIdx.x] += red[threadIdx.x + sft];
        __syncthreads();
    }
    if (threadIdx.x == 0) partial[blockIdx.x] = red[0];
}

// ---------------------------------------------------------------------------
// Kernel 2: per-batch-item sparse corrections (128 threads, one per b).
//  - anchor IoUs -> best anchor (first-max, matches jnp.argmax)
//  - remove default noobj term where reference zeroes noobj
//  - add obj conf BCE, L1 box losses, class BCE at the best cell
//  - idle lanes also zero the reduction padding slots [NPART, NPAD)
// ---------------------------------------------------------------------------
__global__ void yolo_corr(const float* __restrict__ x,
                          const float* __restrict__ labels,
                          float* __restrict__ partial) {
    __shared__ float red[NB];
    const int b = threadIdx.x;

    // zero the padding so the WMMA reduction can run guard-free over NPAD
    if (b < NPAD - NPART) partial[NPART + b] = 0.0f;

    const float* lab = labels + b * 5;
    const float gx = lab[0] * (float)NG;
    const float gy = lab[1] * (float)NG;
    const float gw = lab[2] * (float)NG;
    const float gh = lab[3] * (float)NG;
    const int gi = (int)gx;
    const int gj = (int)gy;
    const int gcls = (int)lab[4];

    const float AW[NA] = { 96.0f / 8.0f, 79.0f / 8.0f, 81.0f / 8.0f };
    const float AH[NA] = { 230.0f / 8.0f, 186.0f / 8.0f, 133.0f / 8.0f };

    float iou[NA];
    int best = 0;
    float bestv = -1.0f;
#pragma unroll
    for (int aa = 0; aa < NA; ++aa) {
        const float inter = fminf(AW[aa], gw) * fminf(AH[aa], gh);
        iou[aa] = inter / (AW[aa] * AH[aa] + 1e-16f + gw * gh - inter);
        if (iou[aa] > bestv) { bestv = iou[aa]; best = aa; }  // first max wins
    }

    const int pix = gj * NG + gi;
    float corr = 0.0f;

    // Undo the default noobj term wherever the reference zeroes noobj:
    // best anchor (set 0) and anchors with iou > IGNORE_THRES (mul 1-ign).
#pragma unroll
    for (int aa = 0; aa < NA; ++aa) {
        const bool removed = (aa == best) || (iou[aa] > 0.5f);
        if (removed) {
            const float cf = sigmoidf_(x[(size_t)(b * 60 + aa * 20 + 4) * PIX + pix]);
            corr += 100.0f * fmaxf(logf(1.0f - cf), -100.0f);  // subtract default
        }
    }

    // obj-cell terms at (b, best, gj, gi)
    const float* cb = x + (size_t)(b * 60 + best * 20) * PIX + pix;
    float v[20];
#pragma unroll
    for (int c = 0; c < 20; ++c) v[c] = cb[(size_t)c * PIX];

    const float pxv  = sigmoidf_(v[0]);
    const float pyv  = sigmoidf_(v[1]);
    const float conf = sigmoidf_(v[4]);

    corr += fabsf(pxv - (gx - (float)gi));                       // loss_x
    corr += fabsf(pyv - (gy - (float)gj));                       // loss_y
    corr += fabsf(v[2] - logf(gw / AW[best] + 1e-16f));          // loss_w
    corr += fabsf(v[3] - logf(gh / AH[best] + 1e-16f));          // loss_h
    corr += -fmaxf(logf(conf), -100.0f);                         // obj conf BCE

    // class BCE against one-hot(gcls) on the softmax
    float m = v[5];
#pragma unroll
    for (int c = 6; c < 20; ++c) m = fmaxf(m, v[c]);
    float e[NC];
    float s = 0.0f;
#pragma unroll
    for (int c = 0; c < NC; ++c) { e[c] = expf(v[5 + c] - m); s += e[c]; }
    const float invs = 1.0f / s;
#pragma unroll
    for (int c = 0; c < NC; ++c) {
        const float pcl = e[c] * invs;
        if (c == gcls) corr += -fmaxf(logf(pcl), -100.0f);
        else           corr += -fmaxf(logf(1.0f - pcl), -100.0f);
    }

    red[b] = corr;
    __syncthreads();
#pragma unroll
    for (int sft = NB / 2; sft > 0; sft >>= 1) {
        if (b < sft) red[b] += red[b + sft];
        __syncthreads();
    }
    if (b == 0) partial[NBLK] = red[0];
}

// ---------------------------------------------------------------------------
// Kernel 3: deterministic final reduction using V_WMMA_F32_16X16X4_F32.
// One wave32 with EXEC all ones (as WMMA requires), guard-free thanks to the
// zeroed padding. Each iteration feeds 64 contiguous partials as the 16x4
// A-matrix (lane L takes floats [2L, 2L+1] via one global_load_b64 — any
// bijection of the 64 chunk values onto A is valid since we only need their
// sum), B = all-ones 4x16, C accumulates:
//   D[m][n] = sum_k A[m][k] + C[m][n]   (all columns identical).
// Grand total = rows 0..7 (lane 0) + rows 8..15 (lane 16).
// ---------------------------------------------------------------------------
typedef __attribute__((ext_vector_type(2))) float v2f;
typedef __attribute__((ext_vector_type(8))) float v8f;

__global__ void yolo_reduce(const float* __restrict__ partial,
                            float* __restrict__ out) {
    __shared__ float red[32];
    const int lane = threadIdx.x;   // 32 threads = 1 wave

    v8f c = {};
    v2f ones;
    ones.x = 1.0f;
    ones.y = 1.0f;

    const v2f* p2 = (const v2f*)partial;   // ws is >=8B aligned; NPAD % 64 == 0
#pragma unroll 4
    for (int chunk = 0; chunk < NPAD / 64; ++chunk) {
        v2f a = p2[chunk * 32 + lane];     // one global_load_b64 per lane
        // 8 args: (neg_a, A, neg_b, B, c_mod, C, reuse_a, reuse_b)
        c = __builtin_amdgcn_wmma_f32_16x16x4_f32(
                false, a, false, ones, (short)0, c, false, false);
    }

    float s = c[0] + c[1] + c[2] + c[3] + c[4] + c[5] + c[6] + c[7];
    red[lane] = s;
    __syncthreads();
    if (lane == 0) out[0] = red[0] + red[16];  // rows 0..7 + rows 8..15
}

// ---------------------------------------------------------------------------
extern "C" void kernel_launch(void* const* d_in, const int* in_sizes, int n_in,
                              void* d_out, int out_size, void* d_ws, size_t ws_size,
                              hipStream_t stream) {
    (void)in_sizes; (void)n_in; (void)out_size; (void)ws_size;
    const float* x      = (const float*)d_in[0];  // (128, 60, 52, 52) f32
    const float* labels = (const float*)d_in[1];  // (128, 5) f32
    float* out = (float*)d_out;                   // [loss, out(128*8112*20)]
    float* ws  = (float*)d_ws;                    // NPAD partial-sum slots

    yolo_main<<<NBLK, BLK, 0, stream>>>(x, out, ws);
    yolo_corr<<<1, NB, 0, stream>>>(x, labels, ws);
    yolo_reduce<<<1, 32, 0, stream>>>(ws, out);
}